// GraphSage_19207093747736
// MI455X (gfx1250) — compile-verified
//
#include <hip/hip_runtime.h>

typedef __attribute__((ext_vector_type(2))) float v2f;
typedef __attribute__((ext_vector_type(8))) float v8f;

#define TM         32          // output rows per block
#define F          128         // feature width (in == out)
#define KDIM       256         // concat width = 2*F
#define NEIGH      10
#define LDS_STRIDE 260         // 256 + 4 pad -> conflict-free A-frag b64 reads

// One fused GraphSAGE layer:
//   combined[r] = [ feats[self[r]] | mean_j feats[neigh[r][j]] ]  (NaN -> 0.5)
//   out[r]      = relu6( combined[r] @ W^T )        W: [F, KDIM] row-major
__global__ __launch_bounds__(256) void sage_layer(
    const float* __restrict__ feats,      // [*, F]
    const float* __restrict__ W,          // [F, KDIM]
    const int*   __restrict__ self_idx,   // [nrows]
    const int*   __restrict__ neigh_idx,  // [nrows, NEIGH]
    float*       __restrict__ out,        // [nrows, F]
    int nrows)
{
    __shared__ __attribute__((aligned(16))) float comb[TM * LDS_STRIDE];

    const int lane = threadIdx.x & 31;
    const int wave = threadIdx.x >> 5;
    const int row0 = blockIdx.x * TM;

    // ---------------- gather + mean phase (each wave builds 4 rows) ----------
    for (int rr = 0; rr < 4; ++rr) {
        const int r  = wave * 4 + rr;
        const int gr = row0 + r;
        const int c4 = lane * 4;                       // 32 lanes * 4 floats = 128

        const long sidx = (long)self_idx[gr];
        const float4 sf = *(const float4*)(feats + sidx * F + c4);
        *(float4*)(&comb[r * LDS_STRIDE + c4]) = sf;

        float4 acc = make_float4(0.f, 0.f, 0.f, 0.f);
        const int* nbr = neigh_idx + (long)gr * NEIGH;
        #pragma unroll
        for (int j = 0; j < NEIGH; ++j) {
            const long nidx = (long)nbr[j];
            const float4 nf = *(const float4*)(feats + nidx * F + c4);
            acc.x += nf.x; acc.y += nf.y; acc.z += nf.z; acc.w += nf.w;
        }
        const float s = 1.0f / (float)NEIGH;
        acc.x *= s; acc.y *= s; acc.z *= s; acc.w *= s;
        // match reference: where(isnan(agg), 0.5, agg)
        acc.x = (acc.x != acc.x) ? 0.5f : acc.x;
        acc.y = (acc.y != acc.y) ? 0.5f : acc.y;
        acc.z = (acc.z != acc.z) ? 0.5f : acc.z;
        acc.w = (acc.w != acc.w) ? 0.5f : acc.w;
        *(float4*)(&comb[r * LDS_STRIDE + F + c4]) = acc;
    }
    __syncthreads();

    // ---------------- WMMA phase: 2 Mtiles x 8 Ntiles, wave owns t and t+8 ---
    const int halfsel = lane >> 4;      // 0: K pair {0,1}, 1: K pair {2,3}
    const int lrow    = lane & 15;      // M (for A) / N (for B,C,D)

    for (int t = wave; t < 16; t += 8) {
        const int m0 = (t >> 3) * 16;
        const int n0 = (t & 7) * 16;

        v8f c = {};
        const float* arow = &comb[(m0 + lrow) * LDS_STRIDE + 2 * halfsel];
        const float* wrow = W + (long)(n0 + lrow) * KDIM + 2 * halfsel;

        #pragma unroll 8
        for (int k = 0; k < KDIM; k += 4) {
            v2f a = *(const v2f*)(arow + k);   // A[m][k+kk], A[m][k+kk+1]
            v2f b = *(const v2f*)(wrow + k);   // B[k+kk][n] = W[n][k+kk]
            c = __builtin_amdgcn_wmma_f32_16x16x4_f32(
                    false, a, false, b, (short)0, c, false, false);
        }

        // relu6 + store; C/D layout: VGPR r -> M = r (lanes 0-15), r+8 (16-31)
        const int n = n0 + lrow;
        #pragma unroll
        for (int r = 0; r < 8; ++r) {
            const int m = m0 + r + 8 * halfsel;
            float v = c[r];
            v = fminf(fmaxf(v, 0.0f), 6.0f);
            out[(long)(row0 + m) * F + n] = v;
        }
    }
}

extern "C" void kernel_launch(void* const* d_in, const int* in_sizes, int n_in,
                              void* d_out, int out_size, void* d_ws, size_t ws_size,
                              hipStream_t stream) {
    const float* raw   = (const float*)d_in[0];   // [N_NODES, 128]
    const float* W1    = (const float*)d_in[1];   // [128, 256]
    const float* W2    = (const float*)d_in[2];   // [128, 256]
    const int*   u1    = (const int*)  d_in[3];   // [U1]
    const int*   nbr1  = (const int*)  d_in[4];   // [U1, 10]
    const int*   self2 = (const int*)  d_in[5];   // [B]
    const int*   nbr2  = (const int*)  d_in[6];   // [B, 10]

    const int U1 = in_sizes[3];                   // 180224 (multiple of 32)
    const int B  = in_sizes[5];                   // 16384  (multiple of 32)

    float* h1 = (float*)d_ws;                     // [U1, 128] scratch (~88 MB)
    float* h2 = (float*)d_out;                    // [B, 128]

    dim3 blk(256);
    // Layer 1: gather raw features, GEMM with W1 -> h1
    sage_layer<<<dim3((U1 + TM - 1) / TM), blk, 0, stream>>>(raw, W1, u1, nbr1, h1, U1);
    // Layer 2: gather h1 (L2-resident), GEMM with W2 -> h2
    sage_layer<<<dim3((B + TM - 1) / TM), blk, 0, stream>>>(h1, W2, self2, nbr2, h2, B);
}